// VolumeSDFRenderer_22797686408153
// MI455X (gfx1250) — compile-verified
//
#include <hip/hip_runtime.h>
#include <hip/hip_bf16.h>

typedef float v2f __attribute__((ext_vector_type(2)));
typedef float v8f __attribute__((ext_vector_type(8)));

#define ALPHA_C    10.0f
#define INV_BETA   20.0f      // 1/0.05
#define FAR_DELTA  1e10f
#define N_PTS      128
#define LDS_STRIDE 132        // 128 + 4 pad -> conflict-free column access
#define WAVES_PB   4
#define RAYS_PW    16

struct __attribute__((packed, aligned(4))) f3 { float x, y, z; };

__device__ __forceinline__ float laplace_density(float dist) {
    // sd = -dist ; sai = sd<=0 ? 0.5*exp(sd/B) : 1 - 0.5*exp(-sd/B)
    float sd = -dist;
    float e  = __expf(-fabsf(sd) * INV_BETA);
    float sai = (sd <= 0.0f) ? (0.5f * e) : (1.0f - 0.5f * e);
    return ALPHA_C * sai;
}

__global__ __launch_bounds__(WAVES_PB * 32, 1)
void volsdf_wmma_kernel(const float* __restrict__ distance,
                        const float* __restrict__ color,
                        const float* __restrict__ depth,
                        float* __restrict__ out)
{
    __shared__ float dd_s[WAVES_PB * RAYS_PW * LDS_STRIDE];

    const int lane    = threadIdx.x & 31;
    const int wave    = threadIdx.x >> 5;
    const int rayBase = (blockIdx.x * WAVES_PB + wave) * RAYS_PW;
    float* dd_w = dd_s + wave * (RAYS_PW * LDS_STRIDE);

    // ---------------- Phase 1: dd = density*delta -> LDS ----------------
    // Iteration t: the wave covers ray (rayBase+t), lanes cover p = 4*lane..4*lane+3.
    const int p0 = lane << 2;
    for (int t = 0; t < RAYS_PW; ++t) {
        const size_t g = (size_t)(rayBase + t) * N_PTS + (size_t)p0;
        const float4 di = *(const float4*)(distance + g);
        const float4 de = *(const float4*)(depth + g);
        float deNext = 0.0f;
        if (lane != 31) deNext = depth[g + 4];   // p0 <= 120 -> in-bounds, same ray

        float4 delta;
        delta.x = de.y - de.x;
        delta.y = de.z - de.y;
        delta.z = de.w - de.z;
        delta.w = (lane == 31) ? FAR_DELTA : (deNext - de.w);

        float4 dd;
        dd.x = laplace_density(di.x) * delta.x;
        dd.y = laplace_density(di.y) * delta.y;
        dd.z = laplace_density(di.z) * delta.z;
        dd.w = laplace_density(di.w) * delta.w;

        *(float4*)(dd_w + t * LDS_STRIDE + p0) = dd;   // 16B-aligned (528*t + 16*lane)
    }
    __syncthreads();

    // ------- Phase 2+3: WMMA exclusive cumsum + weights + color accum -------
    // A (16x4 f32): lane&15 = row M (ray), half selects K pair {0,1} or {2,3}.
    // B (4x16 f32): lane&15 = col N (point), half selects same K pair.
    // C/D (16x16 f32): VGPR v -> M = v + 8*half, N = lane&15.
    const int row  = lane & 15;
    const int half = lane >> 4;
    const int kl0  = half << 1;

    // B masks: full tiles (kg entirely below the output columns) are all-ones.
    // The diagonal tile's mask depends only on (row, half) -> precompute 4 v2f's.
    const v2f bOnes = {1.0f, 1.0f};
    v2f bt[4];
#pragma unroll
    for (int j = 0; j < 4; ++j) {
        const int kgl = (j << 2) + kl0;          // local K within diagonal tile
        bt[j].x = (kgl     < row) ? 1.0f : 0.0f; // strictly-lower-triangular ones
        bt[j].y = (kgl + 1 < row) ? 1.0f : 0.0f;
    }

    float accR[8], accG[8], accB[8];
#pragma unroll
    for (int v = 0; v < 8; ++v) { accR[v] = 0.f; accG[v] = 0.f; accB[v] = 0.f; }

    const float* aRow = dd_w + row * LDS_STRIDE; // this lane's A row base

    for (int nt = 0; nt < 8; ++nt) {            // 16-point output tiles
        v8f acc = {0.f, 0.f, 0.f, 0.f, 0.f, 0.f, 0.f, 0.f};
        // Full tiles: B == all-ones (every K strictly below every output column).
        const int kcFull = nt << 2;
        for (int kc = 0; kc < kcFull; ++kc) {
            v2f a = *(const v2f*)(aRow + (kc << 2) + kl0);
            acc = __builtin_amdgcn_wmma_f32_16x16x4_f32(
                      false, a, false, bOnes, (short)0, acc, false, false);
        }
        // Diagonal tile: 4 chunks with the precomputed triangular masks.
#pragma unroll
        for (int j = 0; j < 4; ++j) {
            v2f a = *(const v2f*)(aRow + ((kcFull + j) << 2) + kl0);
            acc = __builtin_amdgcn_wmma_f32_16x16x4_f32(
                      false, a, false, bt[j], (short)0, acc, false, false);
        }

        // weights + weighted color for this tile; lane owns point p, rays (8*half+v)
        const int p = (nt << 4) + row;
#pragma unroll
        for (int v = 0; v < 8; ++v) {
            const int rl  = (half << 3) + v;
            const float S   = acc[v];                         // exclusive cumsum
            const float ddv = dd_w[rl * LDS_STRIDE + p];
            const float w   = __expf(-S) * (1.0f - __expf(-ddv));
            const size_t ci = ((size_t)(rayBase + rl) * N_PTS + (size_t)p) * 3;
            const f3 c3 = *(const f3*)(color + ci);           // 12B packed load
            accR[v] += w * c3.x;
            accG[v] += w * c3.y;
            accB[v] += w * c3.z;
        }
    }

    // ---- Reduce over the 16 lanes of each half (points), then store ----
#pragma unroll
    for (int v = 0; v < 8; ++v) {
#pragma unroll
        for (int off = 1; off < 16; off <<= 1) {
            accR[v] += __shfl_xor(accR[v], off);
            accG[v] += __shfl_xor(accG[v], off);
            accB[v] += __shfl_xor(accB[v], off);
        }
    }
    if (row == 0) {                 // lanes 0 and 16 each hold 8 finished rays
#pragma unroll
        for (int v = 0; v < 8; ++v) {
            const int ray = rayBase + (half << 3) + v;
            out[ray * 3 + 0] = accR[v];
            out[ray * 3 + 1] = accG[v];
            out[ray * 3 + 2] = accB[v];
        }
    }
}

extern "C" void kernel_launch(void* const* d_in, const int* in_sizes, int n_in,
                              void* d_out, int out_size, void* d_ws, size_t ws_size,
                              hipStream_t stream) {
    const float* distance = (const float*)d_in[0];   // [65536, 128]
    const float* color    = (const float*)d_in[1];   // [65536, 128, 3]
    const float* depth    = (const float*)d_in[2];   // [65536, 128]
    float* out            = (float*)d_out;           // [65536, 3]

    const int n_rays = in_sizes[0] / N_PTS;                       // 65536
    const int blocks = n_rays / (WAVES_PB * RAYS_PW);             // 1024
    volsdf_wmma_kernel<<<blocks, WAVES_PB * 32, 0, stream>>>(distance, color, depth, out);
}